// CustomGATHeadLayer_84172769067966
// MI455X (gfx1250) — compile-verified
//
#include <hip/hip_runtime.h>

#define NN 50000
#define EE 800000
#define DD 128

__device__ __constant__ const float C_EPS    = 1e-6f;
__device__ __constant__ const float C_BNEPS  = 1e-5f;

typedef float v2f __attribute__((ext_vector_type(2)));
typedef float v8f __attribute__((ext_vector_type(8)));

// ---------------------------------------------------------------------------
// Kernel 1: z = h @ fc_w^T using V_WMMA_F32_16X16X4_F32 (full f32 precision;
// GEMM is HBM-bound at 23.3 TB/s so no precision downcast is warranted).
// Block = 256 threads = 8 waves. blockIdx.x = 16-row tile, wave = 16-col tile.
// 50000 = 3125*16 exactly -> no bounds checks, EXEC all-ones for WMMA.
// A (16x4 f32): lane L holds row L%16, K = kBase + 2*(L/16) + {0,1}
// B (4x16 f32): lane L holds col L%16, K = kBase + 2*(L/16) + {0,1}
// C/D (16x16 f32): VGPR r -> row r + 8*(L/16), col L%16
// ---------------------------------------------------------------------------
__global__ void __launch_bounds__(256)
gat_gemm_z(const float* __restrict__ h, const float* __restrict__ w,
           float* __restrict__ z)
{
    const int wave = threadIdx.x >> 5;
    const int lane = threadIdx.x & 31;
    const int half = lane >> 4;
    const int l16  = lane & 15;
    const int rowBase = blockIdx.x << 4;
    const int colBase = wave << 4;

    const float* __restrict__ ha = h + (size_t)(rowBase + l16) * DD + 2 * half;
    const float* __restrict__ wb = w + (size_t)(colBase + l16) * DD + 2 * half;

    v8f acc = {};
#pragma unroll
    for (int k = 0; k < DD; k += 4) {
        v2f a = *(const v2f*)(ha + k);   // global_load_b64
        v2f b = *(const v2f*)(wb + k);   // global_load_b64
        acc = __builtin_amdgcn_wmma_f32_16x16x4_f32(
            false, a, false, b, (short)0, acc, false, false);
    }

    float* __restrict__ zt = z + (size_t)(rowBase + half * 8) * DD + colBase + l16;
#pragma unroll
    for (int r = 0; r < 8; ++r)
        zt[(size_t)r * DD] = acc[r];
}

// ---------------------------------------------------------------------------
// Kernel 2: s1[n] = z[n]·a_src, s2[n] = z[n]·a_dst. One wave32 per node.
// ---------------------------------------------------------------------------
__global__ void __launch_bounds__(256)
gat_s1s2(const float* __restrict__ z, const float* __restrict__ attn_w,
         float* __restrict__ s1, float* __restrict__ s2)
{
    const int wave = threadIdx.x >> 5;
    const int lane = threadIdx.x & 31;
    const int node = blockIdx.x * 8 + wave;
    if (node >= NN) return;

    const int f = lane * 4;
    const float4 zv = *(const float4*)(z + (size_t)node * DD + f);
    const float4 as = *(const float4*)(attn_w + f);
    const float4 ad = *(const float4*)(attn_w + DD + f);

    float d1 = zv.x * as.x + zv.y * as.y + zv.z * as.z + zv.w * as.w;
    float d2 = zv.x * ad.x + zv.y * ad.y + zv.z * ad.z + zv.w * ad.w;
#pragma unroll
    for (int off = 16; off > 0; off >>= 1) {
        d1 += __shfl_down(d1, off, 32);
        d2 += __shfl_down(d2, off, 32);
    }
    if (lane == 0) { s1[node] = d1; s2[node] = d2; }
}

// ---------------------------------------------------------------------------
// Kernel 3: per-edge leaky-relu score + segment-max via monotone-uint atomicMax
// ---------------------------------------------------------------------------
__device__ __forceinline__ unsigned fmax_encode(float f) {
    unsigned b = __float_as_uint(f);
    return (b & 0x80000000u) ? ~b : (b | 0x80000000u);
}
__device__ __forceinline__ float fmax_decode(unsigned u) {
    unsigned b = (u & 0x80000000u) ? (u ^ 0x80000000u) : ~u;
    return __uint_as_float(b);
}

__global__ void __launch_bounds__(256)
gat_edge_max(const int* __restrict__ src, const int* __restrict__ dst,
             const float* __restrict__ s1, const float* __restrict__ s2,
             float* __restrict__ ebuf, unsigned* __restrict__ emax_u)
{
    const int i = blockIdx.x * 256 + threadIdx.x;
    if (i >= EE) return;
    float e = s1[src[i]] + s2[dst[i]];
    e = (e > 0.0f) ? e : 0.01f * e;        // leaky_relu slope 0.01
    ebuf[i] = e;
    atomicMax(emax_u + dst[i], fmax_encode(e));
}

// ---------------------------------------------------------------------------
// Kernel 4: num = exp(e - emax[dst]); den[dst] += num
// ---------------------------------------------------------------------------
__global__ void __launch_bounds__(256)
gat_edge_exp(const int* __restrict__ dst, float* __restrict__ ebuf,
             const unsigned* __restrict__ emax_u, float* __restrict__ den)
{
    const int i = blockIdx.x * 256 + threadIdx.x;
    if (i >= EE) return;
    const int d = dst[i];
    const float num = __expf(ebuf[i] - fmax_decode(emax_u[d]));
    ebuf[i] = num;
    atomicAdd(den + d, num);
}

// ---------------------------------------------------------------------------
// Kernel 5 (dominant scatter): one wave32 per edge, 4 feats/lane.
// s[dst][f] += |alpha*z[src][f] + EPS| ^ clip(p[f],1,100)
// z[src] gather + atomics ~0.9 GB; s (25.6 MB) lives in the 192 MB L2.
// ---------------------------------------------------------------------------
__global__ void __launch_bounds__(256)
gat_edge_scatter(const int* __restrict__ src, const int* __restrict__ dst,
                 const float* __restrict__ ebuf, const float* __restrict__ den,
                 const float* __restrict__ z, const float* __restrict__ p,
                 float* __restrict__ s)
{
    const int wave = threadIdx.x >> 5;
    const int lane = threadIdx.x & 31;
    const int i = blockIdx.x * 8 + wave;
    if (i >= EE) return;

    const int sn = src[i], dn = dst[i];
    const float alpha = ebuf[i] / den[dn];

    const int f = lane * 4;
    const float4 zv = *(const float4*)(z + (size_t)sn * DD + f);
    const float4 pv = *(const float4*)(p + f);

    const float p0 = fminf(fmaxf(pv.x, 1.0f), 100.0f);
    const float p1 = fminf(fmaxf(pv.y, 1.0f), 100.0f);
    const float p2 = fminf(fmaxf(pv.z, 1.0f), 100.0f);
    const float p3 = fminf(fmaxf(pv.w, 1.0f), 100.0f);

    const float c0 = __powf(fabsf(alpha * zv.x + C_EPS), p0);
    const float c1 = __powf(fabsf(alpha * zv.y + C_EPS), p1);
    const float c2 = __powf(fabsf(alpha * zv.z + C_EPS), p2);
    const float c3 = __powf(fabsf(alpha * zv.w + C_EPS), p3);

    float* sp = s + (size_t)dn * DD + f;
    atomicAdd(sp + 0, c0);
    atomicAdd(sp + 1, c1);
    atomicAdd(sp + 2, c2);
    atomicAdd(sp + 3, c3);
}

// ---------------------------------------------------------------------------
// Kernel 6: hn = (s+EPS)^(1/p) -> d_out (staging); per-feature sum/sumsq
// via register accumulation -> LDS reduce -> 128 global atomics per block.
// ---------------------------------------------------------------------------
__global__ void __launch_bounds__(256)
gat_hn_stats(const float* __restrict__ s, const float* __restrict__ p,
             float* __restrict__ hn, float* __restrict__ fsum,
             float* __restrict__ fsq)
{
    __shared__ float bsum[DD];
    __shared__ float bsq[DD];
    const int wave = threadIdx.x >> 5;
    const int lane = threadIdx.x & 31;
    const int f = lane * 4;

    if (threadIdx.x < DD) { bsum[threadIdx.x] = 0.0f; bsq[threadIdx.x] = 0.0f; }
    __syncthreads();

    const float4 pv = *(const float4*)(p + f);
    const float i0 = 1.0f / fminf(fmaxf(pv.x, 1.0f), 100.0f);
    const float i1 = 1.0f / fminf(fmaxf(pv.y, 1.0f), 100.0f);
    const float i2 = 1.0f / fminf(fmaxf(pv.z, 1.0f), 100.0f);
    const float i3 = 1.0f / fminf(fmaxf(pv.w, 1.0f), 100.0f);

    float as0 = 0, as1 = 0, as2 = 0, as3 = 0;
    float aq0 = 0, aq1 = 0, aq2 = 0, aq3 = 0;

    for (int node = blockIdx.x * 8 + wave; node < NN; node += gridDim.x * 8) {
        const float4 sv = *(const float4*)(s + (size_t)node * DD + f);
        const float h0 = __powf(sv.x + C_EPS, i0);
        const float h1 = __powf(sv.y + C_EPS, i1);
        const float h2 = __powf(sv.z + C_EPS, i2);
        const float h3 = __powf(sv.w + C_EPS, i3);
        *(float4*)(hn + (size_t)node * DD + f) = make_float4(h0, h1, h2, h3);
        as0 += h0; as1 += h1; as2 += h2; as3 += h3;
        aq0 += h0 * h0; aq1 += h1 * h1; aq2 += h2 * h2; aq3 += h3 * h3;
    }

    atomicAdd(&bsum[f + 0], as0); atomicAdd(&bsq[f + 0], aq0);
    atomicAdd(&bsum[f + 1], as1); atomicAdd(&bsq[f + 1], aq1);
    atomicAdd(&bsum[f + 2], as2); atomicAdd(&bsq[f + 2], aq2);
    atomicAdd(&bsum[f + 3], as3); atomicAdd(&bsq[f + 3], aq3);
    __syncthreads();

    if (threadIdx.x < DD) {
        atomicAdd(fsum + threadIdx.x, bsum[threadIdx.x]);
        atomicAdd(fsq  + threadIdx.x, bsq[threadIdx.x]);
    }
}

// ---------------------------------------------------------------------------
// Kernel 7: BatchNorm (mean/var from sums) + ELU, in place on d_out.
// ---------------------------------------------------------------------------
__global__ void __launch_bounds__(256)
gat_bn_elu(const float* __restrict__ fsum, const float* __restrict__ fsq,
           const float* __restrict__ gamma, const float* __restrict__ beta,
           float* __restrict__ out)
{
    const size_t idx = (size_t)blockIdx.x * 256 + threadIdx.x;
    if (idx >= (size_t)NN * DD) return;
    const int f = (int)(idx & (DD - 1));
    const float invN = 1.0f / (float)NN;
    const float mean = fsum[f] * invN;
    const float var  = fsq[f] * invN - mean * mean;
    float x = out[idx];
    x = (x - mean) * rsqrtf(var + C_BNEPS) * gamma[f] + beta[f];
    out[idx] = (x > 0.0f) ? x : (__expf(x) - 1.0f);   // ELU alpha=1
}

// ---------------------------------------------------------------------------
// Host launcher. Workspace layout (floats):
//   z[6.4M] | s[6.4M] | den[50k] | emax_u[50k] | fsum[128] | fsq[128]
//   | ebuf[800k] | s1[50k] | s2[50k]   (~55.2 MB total)
// The [s .. fsq] range is contiguous and zeroed via hipMemsetAsync.
// ---------------------------------------------------------------------------
extern "C" void kernel_launch(void* const* d_in, const int* in_sizes, int n_in,
                              void* d_out, int out_size, void* d_ws, size_t ws_size,
                              hipStream_t stream)
{
    const float* h      = (const float*)d_in[0];
    const int*   src    = (const int*)d_in[1];
    const int*   dst    = (const int*)d_in[2];
    const float* fc_w   = (const float*)d_in[3];
    const float* attn_w = (const float*)d_in[4];
    const float* p      = (const float*)d_in[5];
    const float* gamma  = (const float*)d_in[6];
    const float* beta   = (const float*)d_in[7];
    float* out = (float*)d_out;

    float* ws = (float*)d_ws;
    float*    z      = ws;                                   // 6,400,000
    float*    s      = z + (size_t)NN * DD;                  // 6,400,000
    float*    den    = s + (size_t)NN * DD;                  // 50,000
    unsigned* emax_u = (unsigned*)(den + NN);                // 50,000
    float*    fsum   = (float*)(emax_u + NN);                // 128
    float*    fsq    = fsum + DD;                            // 128
    float*    ebuf   = fsq + DD;                             // 800,000
    float*    s1     = ebuf + EE;                            // 50,000
    float*    s2     = s1 + NN;                              // 50,000

    // Zero s, den, emax_u, fsum, fsq (contiguous region).
    const size_t zero_bytes = ((size_t)NN * DD + 2 * NN + 2 * DD) * sizeof(float);
    hipMemsetAsync(s, 0, zero_bytes, stream);

    gat_gemm_z<<<NN / 16, 256, 0, stream>>>(h, fc_w, z);
    gat_s1s2<<<NN / 8, 256, 0, stream>>>(z, attn_w, s1, s2);
    gat_edge_max<<<(EE + 255) / 256, 256, 0, stream>>>(src, dst, s1, s2, ebuf, emax_u);
    gat_edge_exp<<<(EE + 255) / 256, 256, 0, stream>>>(dst, ebuf, emax_u, den);
    gat_edge_scatter<<<EE / 8, 256, 0, stream>>>(src, dst, ebuf, den, z, p, s);
    gat_hn_stats<<<256, 256, 0, stream>>>(s, p, out, fsum, fsq);
    gat_bn_elu<<<(NN * DD + 255) / 256, 256, 0, stream>>>(fsum, fsq, gamma, beta, out);
}